// GraphConvEncoder_30124900614325
// MI455X (gfx1250) — compile-verified
//
#include <hip/hip_runtime.h>
#include <hip/hip_bf16.h>

typedef __attribute__((ext_vector_type(2))) float v2f;
typedef __attribute__((ext_vector_type(8))) float v8f;

#define DH 128

__device__ __forceinline__ void atomAddF(float* p, float v) {
    __hip_atomic_fetch_add(p, v, __ATOMIC_RELAXED, __HIP_MEMORY_SCOPE_AGENT);
}

// ---------------------------------------------------------------------------
// Edge scatter: agg[dst[e], :] += w[e] * x[src[e], :]
// One thread per (edge, 4-feature group). groups = d/4, gshift = log2(groups).
// ---------------------------------------------------------------------------
__global__ void scatter_edges(const float* __restrict__ x,
                              const int* __restrict__ src,
                              const int* __restrict__ dst,
                              const float* __restrict__ w,
                              float* __restrict__ agg,
                              int E, int d, int gshift) {
    int tid = blockIdx.x * blockDim.x + threadIdx.x;
    int gmask = (1 << gshift) - 1;
    int e = tid >> gshift;
    int g = tid & gmask;
    if (e >= E) return;
    int s = src[e];
    int t = dst[e];
    float wt = w[e];
    const float4 xv = *(const float4*)(x + (size_t)s * d + g * 4);
    float* ap = agg + (size_t)t * d + g * 4;
    atomAddF(ap + 0, xv.x * wt);
    atomAddF(ap + 1, xv.y * wt);
    atomAddF(ap + 2, xv.z * wt);
    atomAddF(ap + 3, xv.w * wt);
}

// ---------------------------------------------------------------------------
// WMMA GEMM: Hout[N,128] = A0[N,K] @ W0[K,128] + A1[N,K] @ W1[K,128] + bias
// One wave per 16x16 output tile, exact f32 via V_WMMA_F32_16X16X4_F32.
// A layout (16x4 f32): lane l holds row m=l%16, K elements {2*(l/16), 2*(l/16)+1}
// B layout (4x16 f32): VGPR j, lane l holds B[2*(l/16)+j][l%16]
// C/D layout (16x16 f32): VGPR r, lane l holds D[r + 8*(l/16)][l%16]
// ---------------------------------------------------------------------------
__global__ __launch_bounds__(32)
void gemm_wmma_f32(const float* __restrict__ A0, const float* __restrict__ W0,
                   const float* __restrict__ A1, const float* __restrict__ W1,
                   const float* __restrict__ bias, float* __restrict__ Hout,
                   int Nrows, int K) {
    const int lane   = threadIdx.x & 31;
    const int half   = lane >> 4;       // 0 or 1
    const int mloc   = lane & 15;
    const int rowBase = blockIdx.x * 16;
    const int colBase = blockIdx.y * 16;
    const int col    = colBase + mloc;  // N-index for B/C/D
    int m = rowBase + mloc;             // row for A loads
    if (m >= Nrows) m = Nrows - 1;      // clamp (N divisible by 16 anyway)

    v8f acc;
    const float bv = bias[col];
#pragma unroll
    for (int r = 0; r < 8; ++r) acc[r] = bv;

    const float* As[2] = {A0, A1};
    const float* Ws[2] = {W0, W1};
#pragma unroll
    for (int p = 0; p < 2; ++p) {
        const float* __restrict__ arow = As[p] + (size_t)m * K;
        const float* __restrict__ W    = Ws[p];
        for (int k0 = 0; k0 < K; k0 += 4) {
            const int ka = k0 + 2 * half;
            v2f a, b;
            a.x = arow[ka];
            a.y = arow[ka + 1];
            b.x = W[(size_t)ka * DH + col];
            b.y = W[(size_t)(ka + 1) * DH + col];
            acc = __builtin_amdgcn_wmma_f32_16x16x4_f32(
                /*neg_a=*/false, a, /*neg_b=*/false, b,
                /*c_mod=*/(short)0, acc, /*reuse_a=*/false, /*reuse_b=*/false);
        }
    }

    // store D
    float* out = Hout + (size_t)(rowBase + 8 * half) * DH + col;
#pragma unroll
    for (int r = 0; r < 8; ++r) {
        int mo = rowBase + 8 * half + r;
        if (mo < Nrows) out[(size_t)r * DH] = acc[r];
    }
}

// ---------------------------------------------------------------------------
// BatchNorm statistics: one block per feature, deterministic LDS reduction.
// Writes mu[f] and inv[f] = rsqrt(var + eps).
// ---------------------------------------------------------------------------
__global__ __launch_bounds__(256)
void bn_stats(const float* __restrict__ H, float* __restrict__ mu,
              float* __restrict__ inv, int Nrows) {
    const int f = blockIdx.x;            // 0..127
    float s = 0.f, s2 = 0.f;
    for (int n = threadIdx.x; n < Nrows; n += blockDim.x) {
        float v = H[(size_t)n * DH + f];
        s += v;
        s2 += v * v;
    }
    __shared__ float sh[256];
    __shared__ float sh2[256];
    sh[threadIdx.x] = s;
    sh2[threadIdx.x] = s2;
    __syncthreads();
    for (int st = 128; st > 0; st >>= 1) {
        if (threadIdx.x < st) {
            sh[threadIdx.x] += sh[threadIdx.x + st];
            sh2[threadIdx.x] += sh2[threadIdx.x + st];
        }
        __syncthreads();
    }
    if (threadIdx.x == 0) {
        float m   = sh[0] / (float)Nrows;
        float var = sh2[0] / (float)Nrows - m * m;
        mu[f]  = m;
        inv[f] = rsqrtf(var + 1e-5f);
    }
}

// ---------------------------------------------------------------------------
// Fused BN + ReLU (in place)
// ---------------------------------------------------------------------------
__global__ void bn_relu(float* __restrict__ H, const float* __restrict__ mu,
                        const float* __restrict__ inv,
                        const float* __restrict__ gamma,
                        const float* __restrict__ beta, int total) {
    int t = blockIdx.x * blockDim.x + threadIdx.x;
    if (t >= total) return;
    int f = t & (DH - 1);
    float v = (H[t] - mu[f]) * inv[f] * gamma[f] + beta[f];
    H[t] = v > 0.f ? v : 0.f;
}

// ---------------------------------------------------------------------------
// Global mean pool: accumulate sums per graph + counts, then divide.
// ---------------------------------------------------------------------------
__global__ void pool_accum(const float* __restrict__ H,
                           const int* __restrict__ batch,
                           float* __restrict__ psum, float* __restrict__ cnt,
                           int total) {
    int t = blockIdx.x * blockDim.x + threadIdx.x;
    if (t >= total) return;
    int n = t >> 7;          // /128
    int f = t & (DH - 1);
    int g = batch[n];
    atomAddF(psum + (size_t)g * DH + f, H[t]);
    if (f == 0) atomAddF(cnt + g, 1.0f);
}

__global__ void pool_finalize(const float* __restrict__ psum,
                              const float* __restrict__ cnt,
                              float* __restrict__ out, int total) {
    int t = blockIdx.x * blockDim.x + threadIdx.x;
    if (t >= total) return;
    int g = t >> 7;
    float c = cnt[g];
    out[t] = psum[t] / (c > 1.0f ? c : 1.0f);
}

// ---------------------------------------------------------------------------
extern "C" void kernel_launch(void* const* d_in, const int* in_sizes, int n_in,
                              void* d_out, int out_size, void* d_ws, size_t ws_size,
                              hipStream_t stream) {
    const float* x      = (const float*)d_in[0];
    const int*   eidx   = (const int*)d_in[1];
    const float* ew     = (const float*)d_in[2];
    const int*   batch  = (const int*)d_in[3];
    const float* W_rel0 = (const float*)d_in[4];
    const float* b0     = (const float*)d_in[5];
    const float* W_root0= (const float*)d_in[6];
    const float* W_rel1 = (const float*)d_in[7];
    const float* b1     = (const float*)d_in[8];
    const float* W_root1= (const float*)d_in[9];
    const float* W_rel2 = (const float*)d_in[10];
    const float* b2     = (const float*)d_in[11];
    const float* W_root2= (const float*)d_in[12];
    const float* gamma0 = (const float*)d_in[13];
    const float* beta0  = (const float*)d_in[14];
    const float* gamma1 = (const float*)d_in[15];
    const float* beta1  = (const float*)d_in[16];

    const int N   = in_sizes[3];           // batch has N entries
    const int E   = in_sizes[2];           // edge_weight has E entries
    const int DIN = in_sizes[0] / N;       // 64
    const int G   = out_size / DH;         // 256
    const int* src = eidx;
    const int* dst = eidx + E;

    // workspace layout (floats)
    float* agg  = (float*)d_ws;                       // N*DH
    float* H0   = agg + (size_t)N * DH;               // N*DH
    float* H1   = H0  + (size_t)N * DH;               // N*DH
    float* mu   = H1  + (size_t)N * DH;               // DH
    float* inv  = mu + DH;                            // DH
    float* psum = inv + DH;                           // G*DH
    float* cnt  = psum + (size_t)G * DH;              // G
    (void)ws_size; (void)n_in;

    const int TB = 256;
    const int totND  = N * DH;
    const int gridND = (totND + TB - 1) / TB;
    const dim3 gemmGrid((N + 15) / 16, DH / 16);

    // ---------------- layer 0 (d = DIN = 64) ----------------
    hipMemsetAsync(agg, 0, (size_t)N * DIN * sizeof(float), stream);
    {
        int groups = DIN / 4, gshift = 4;              // 16 groups
        long long tot = (long long)E * groups;
        int blocks = (int)((tot + TB - 1) / TB);
        scatter_edges<<<blocks, TB, 0, stream>>>(x, src, dst, ew, agg, E, DIN, gshift);
    }
    gemm_wmma_f32<<<gemmGrid, 32, 0, stream>>>(agg, W_rel0, x, W_root0, b0, H0, N, DIN);
    bn_stats<<<DH, 256, 0, stream>>>(H0, mu, inv, N);
    bn_relu<<<gridND, TB, 0, stream>>>(H0, mu, inv, gamma0, beta0, totND);

    // ---------------- layer 1 (d = DH = 128) ----------------
    hipMemsetAsync(agg, 0, (size_t)N * DH * sizeof(float), stream);
    {
        int groups = DH / 4, gshift = 5;               // 32 groups
        long long tot = (long long)E * groups;
        int blocks = (int)((tot + TB - 1) / TB);
        scatter_edges<<<blocks, TB, 0, stream>>>(H0, src, dst, ew, agg, E, DH, gshift);
    }
    gemm_wmma_f32<<<gemmGrid, 32, 0, stream>>>(agg, W_rel1, H0, W_root1, b1, H1, N, DH);
    bn_stats<<<DH, 256, 0, stream>>>(H1, mu, inv, N);
    bn_relu<<<gridND, TB, 0, stream>>>(H1, mu, inv, gamma1, beta1, totND);

    // ---------------- layer 2 (no BN/ReLU) ----------------
    hipMemsetAsync(agg, 0, (size_t)N * DH * sizeof(float), stream);
    {
        int groups = DH / 4, gshift = 5;
        long long tot = (long long)E * groups;
        int blocks = (int)((tot + TB - 1) / TB);
        scatter_edges<<<blocks, TB, 0, stream>>>(H1, src, dst, ew, agg, E, DH, gshift);
    }
    gemm_wmma_f32<<<gemmGrid, 32, 0, stream>>>(agg, W_rel2, H1, W_root2, b2, H0, N, DH);

    // ---------------- global mean pool ----------------
    hipMemsetAsync(psum, 0, ((size_t)G * DH + G) * sizeof(float), stream);
    pool_accum<<<gridND, TB, 0, stream>>>(H0, batch, psum, cnt, totND);
    int gridG = (G * DH + TB - 1) / TB;
    pool_finalize<<<gridG, TB, 0, stream>>>(psum, cnt, (float*)d_out, G * DH);
}